// Net_84275848282732
// MI455X (gfx1250) — compile-verified
//
#include <hip/hip_runtime.h>
#include <hip/hip_bf16.h>
#include <cstdint>
#include <cstddef>

#define NNODES 20000
#define NEDGES 100000
#define IN_F   167
#define HID    1024
#define EA_STRIDE 12   // edge_attr width = ED + 2
#define KC 64          // K-chunk held in LDS for the WMMA GEMM

typedef __attribute__((ext_vector_type(2))) float v2f;
typedef __attribute__((ext_vector_type(8))) float v8f;

// ---------- order-preserving float<->uint encoding for atomic max ----------
__device__ __forceinline__ unsigned enc_f32(float f) {
  unsigned u = __float_as_uint(f);
  return (u & 0x80000000u) ? ~u : (u | 0x80000000u);
}
__device__ __forceinline__ float dec_f32(unsigned e) {
  unsigned u = (e & 0x80000000u) ? (e & 0x7FFFFFFFu) : ~e;
  return __uint_as_float(u);
}
#define ENC_NEG_INF 0x007FFFFFu   // enc(-inf)

// ------------------------------- fills -------------------------------------
__global__ void fill_f32_k(float* p, float v, size_t n) {
  size_t i = (size_t)blockIdx.x * blockDim.x + threadIdx.x;
  if (i < n) p[i] = v;
}
__global__ void fill_u32_k(unsigned* p, unsigned v, size_t n) {
  size_t i = (size_t)blockIdx.x * blockDim.x + threadIdx.x;
  if (i < n) p[i] = v;
}

// ------------------------------ BatchNorm -----------------------------------
__global__ void bn_reduce_k(const float* __restrict__ X, float* __restrict__ sum,
                            float* __restrict__ sumsq, int nrows, int C) {
  int c = blockIdx.x * blockDim.x + threadIdx.x;
  if (c >= C) return;
  int r0 = blockIdx.y * 128;
  int r1 = r0 + 128; if (r1 > nrows) r1 = nrows;
  float s = 0.f, q = 0.f;
  for (int r = r0; r < r1; ++r) {
    float v = X[(size_t)r * C + c];
    s += v; q += v * v;
  }
  atomicAdd(&sum[c], s);
  atomicAdd(&sumsq[c], q);
}

__global__ void bn_scale_k(const float* __restrict__ sum, const float* __restrict__ sumsq,
                           const float* __restrict__ g, const float* __restrict__ b,
                           float* __restrict__ scale, float* __restrict__ shift,
                           int C, float invN) {
  int c = blockIdx.x * blockDim.x + threadIdx.x;
  if (c >= C) return;
  float mu  = sum[c] * invN;
  float var = sumsq[c] * invN - mu * mu;
  float sc  = rsqrtf(var + 1e-5f) * g[c];
  scale[c] = sc;
  shift[c] = b[c] - mu * sc;
}

__global__ void bn_apply_k(const float* __restrict__ X, const float* __restrict__ scale,
                           const float* __restrict__ shift, float* __restrict__ Y,
                           size_t total, int C) {
  size_t i = (size_t)blockIdx.x * blockDim.x + threadIdx.x;
  if (i >= total) return;
  int c = (int)(i % C);
  Y[i] = X[i] * scale[c] + shift[c];
}

// --------------------------- WMMA f32 GEMM ----------------------------------
// C[nrows x M] = A[nrows x K] @ W[K x M] (+ bias1[col] + bias2[col])
// block = 256 threads = 8 waves; block tile = 16 rows x 128 cols (wave: 16x16)
__global__ void gemm_wmma_k(const float* __restrict__ A, const float* __restrict__ W,
                            const float* __restrict__ bias1, const float* __restrict__ bias2,
                            float* __restrict__ Cout, int nrows, int K, int M) {
  __shared__ float As[16][KC + 4];
  const int wave = threadIdx.x >> 5;
  const int lane = threadIdx.x & 31;
  const int rowBase = blockIdx.y * 16;
  const int colBase = blockIdx.x * 128 + wave * 16;
  const int r16  = lane & 15;
  const int half = lane >> 4;

  v8f acc = {};

  for (int kb = 0; kb < K; kb += KC) {
    // cooperatively stage the 16 x KC A-slab (zero padded past K)
    for (int t = threadIdx.x; t < 16 * KC; t += blockDim.x) {
      int rr = t / KC, kk = t % KC;
      int gk = kb + kk;
      As[rr][kk] = (gk < K) ? A[(size_t)(rowBase + rr) * K + gk] : 0.0f;
    }
    __syncthreads();

    for (int kk = 0; kk < KC; kk += 4) {
      int k0 = kk + half * 2;                  // lanes 0-15: K=kk,kk+1 ; 16-31: kk+2,kk+3
      v2f a;
      a.x = As[r16][k0];
      a.y = As[r16][k0 + 1];
      int gk0 = kb + k0;
      int gk0c = (gk0     < K) ? gk0     : (K - 1);   // A is zero there -> safe clamp
      int gk1c = (gk0 + 1 < K) ? gk0 + 1 : (K - 1);
      v2f b;
      b.x = W[(size_t)gk0c * M + colBase + r16];
      b.y = W[(size_t)gk1c * M + colBase + r16];
      acc = __builtin_amdgcn_wmma_f32_16x16x4_f32(
          /*neg_a=*/false, a, /*neg_b=*/false, b,
          /*c_mod=*/(short)0, acc, /*reuse_a=*/false, /*reuse_b=*/false);
    }
    __syncthreads();
  }

  const int col = colBase + r16;
  float badd = 0.f;
  if (bias1) badd += bias1[col];
  if (bias2) badd += bias2[col];
#pragma unroll
  for (int i = 0; i < 8; ++i) {
    int row = rowBase + i + 8 * half;          // C layout: vgpr i -> M=i (+8 for hi half)
    Cout[(size_t)row * M + col] = acc[i] + badd;
  }
}

// ---------------------- tiny GEMM for M==2 (layer 3) ------------------------
__global__ void gemv2_k(const float* __restrict__ A, const float* __restrict__ W,
                        const float* __restrict__ bias1, const float* __restrict__ bias2,
                        float* __restrict__ out, int nrows, int K) {
  int node = blockIdx.x * (blockDim.x >> 5) + (threadIdx.x >> 5);
  int lane = threadIdx.x & 31;
  if (node >= nrows) return;
  float a0 = 0.f, a1 = 0.f;
  const float* arow = A + (size_t)node * K;
  for (int k = lane; k < K; k += 32) {
    float av = arow[k];
    a0 += av * W[(size_t)k * 2 + 0];
    a1 += av * W[(size_t)k * 2 + 1];
  }
  for (int off = 16; off; off >>= 1) {
    a0 += __shfl_down(a0, off);
    a1 += __shfl_down(a1, off);
  }
  if (lane == 0) {
    float b0 = 0.f, b1v = 0.f;
    if (bias1) { b0 += bias1[0]; b1v += bias1[1]; }
    if (bias2) { b0 += bias2[0]; b1v += bias2[1]; }
    out[(size_t)node * 2 + 0] = a0 + b0;
    out[(size_t)node * 2 + 1] = a1 + b1v;
  }
}

// ----------------------------- GAT pieces -----------------------------------
// one wave per (node, head): s_src = <hg, attn[0,h]>, s_dst = <hg, attn[1,h]>
__global__ void gat_scores_k(const float* __restrict__ hg, const float* __restrict__ attn,
                             float* __restrict__ s_src, float* __restrict__ s_dst,
                             int nrows, int heads, int O) {
  int wid  = blockIdx.x * (blockDim.x >> 5) + (threadIdx.x >> 5);
  int lane = threadIdx.x & 31;
  if (wid >= nrows * heads) return;
  int n = wid / heads, h = wid % heads;
  const float* base = hg + (size_t)n * heads * O + (size_t)h * O;
  const float* a0 = attn + (size_t)h * O;                       // attn[0][h]
  const float* a1 = attn + (size_t)heads * O + (size_t)h * O;   // attn[1][h]
  float s0 = 0.f, s1 = 0.f;
  for (int c = lane; c < O; c += 32) {
    float v = base[c];
    s0 += v * a0[c];
    s1 += v * a1[c];
  }
  for (int off = 16; off; off >>= 1) {
    s0 += __shfl_down(s0, off);
    s1 += __shfl_down(s1, off);
  }
  if (lane == 0) { s_src[wid] = s0; s_dst[wid] = s1; }
}

// proj[d,h] = <We[d, h*O : (h+1)*O], attn[2,h]> ; one wave per (d,h), d<10
__global__ void gat_proj_k(const float* __restrict__ We, const float* __restrict__ attn,
                           float* __restrict__ proj, int heads, int O) {
  int wid  = blockIdx.x * (blockDim.x >> 5) + (threadIdx.x >> 5);
  int lane = threadIdx.x & 31;
  if (wid >= 10 * heads) return;
  int d = wid / heads, h = wid % heads;
  const float* a2 = attn + (size_t)2 * heads * O + (size_t)h * O;
  const float* wr = We + (size_t)d * heads * O + (size_t)h * O;
  float s = 0.f;
  for (int c = lane; c < O; c += 32) s += wr[c] * a2[c];
  for (int off = 16; off; off >>= 1) s += __shfl_down(s, off);
  if (lane == 0) proj[d * heads + h] = s;
}

__global__ void gat_logits_k(const int* __restrict__ ei, const float* __restrict__ ea,
                             const float* __restrict__ s_src, const float* __restrict__ s_dst,
                             const float* __restrict__ proj, const float* __restrict__ at,
                             float* __restrict__ logits, unsigned* __restrict__ m_enc,
                             int nedges, int heads) {
  int i = blockIdx.x * blockDim.x + threadIdx.x;
  if (i >= nedges * heads) return;
  int e = i / heads, h = i % heads;
  int src = ei[e], dst = ei[nedges + e];
  float t = ea[(size_t)e * EA_STRIDE];
  float se = 0.f;
#pragma unroll
  for (int d = 0; d < 10; ++d)
    se += ea[(size_t)e * EA_STRIDE + 1 + d] * proj[d * heads + h];
  float x = s_src[src * heads + h] + s_dst[dst * heads + h] + se + t * at[h];
  float lg = (x > 0.f) ? x : 0.2f * x;               // leaky_relu(0.2)
  logits[i] = lg;
  atomicMax(&m_enc[dst * heads + h], enc_f32(lg));
}

__global__ void gat_softmax_k(const int* __restrict__ ei, const float* __restrict__ logits,
                              const unsigned* __restrict__ m_enc, float* __restrict__ wbuf,
                              float* __restrict__ denom, int nedges, int heads) {
  int i = blockIdx.x * blockDim.x + threadIdx.x;
  if (i >= nedges * heads) return;
  int e = i / heads, h = i % heads;
  int dst = ei[nedges + e];
  float m = dec_f32(m_enc[dst * heads + h]);         // finite: dst has >=1 edge
  float p = expf(logits[i] - m);
  wbuf[i] = p;
  atomicAdd(&denom[dst * heads + h], p);
}

// flattened over (edge, feature j in [0, heads*O)); recompute e-feature on the fly
__global__ void gat_message_k(const int* __restrict__ ei, const float* __restrict__ ea,
                              const float* __restrict__ We, const float* __restrict__ hg,
                              const float* __restrict__ wbuf, const float* __restrict__ denom,
                              float* __restrict__ Cacc, int nedges, int heads, int O) {
  size_t i = (size_t)blockIdx.x * blockDim.x + threadIdx.x;
  int HO = heads * O;
  if (i >= (size_t)nedges * HO) return;
  int e = (int)(i / HO);
  int j = (int)(i % HO);
  int h = j / O;
  int src = ei[e], dst = ei[nedges + e];
  float wn = wbuf[e * heads + h] / (denom[dst * heads + h] + 1e-16f);
  float efv = 0.f;
#pragma unroll
  for (int d = 0; d < 10; ++d)
    efv += ea[(size_t)e * EA_STRIDE + 1 + d] * We[(size_t)d * HO + j];
  float val = wn * (hg[(size_t)src * HO + j] + efv);
  atomicAdd(&Cacc[(size_t)dst * HO + j], val);
}

__global__ void relu_k(float* p, size_t n) {
  size_t i = (size_t)blockIdx.x * blockDim.x + threadIdx.x;
  if (i < n) p[i] = fmaxf(p[i], 0.0f);
}

// ============================== host side ===================================
extern "C" void kernel_launch(void* const* d_in, const int* in_sizes, int n_in,
                              void* d_out, int out_size, void* d_ws, size_t ws_size,
                              hipStream_t stream) {
  (void)in_sizes; (void)n_in; (void)out_size; (void)ws_size;

  const float* x  = (const float*)d_in[0];
  const int*   ei = (const int*)d_in[1];
  const float* ea = (const float*)d_in[2];
  const float* Wg[3]   = {(const float*)d_in[3],  (const float*)d_in[8],  (const float*)d_in[13]};
  const float* We[3]   = {(const float*)d_in[4],  (const float*)d_in[9],  (const float*)d_in[14]};
  const float* attn[3] = {(const float*)d_in[5],  (const float*)d_in[10], (const float*)d_in[15]};
  const float* at[3]   = {(const float*)d_in[6],  (const float*)d_in[11], (const float*)d_in[16]};
  const float* cb[3]   = {(const float*)d_in[7],  (const float*)d_in[12], (const float*)d_in[17]};
  const float* lw[3]   = {(const float*)d_in[18], (const float*)d_in[22], (const float*)d_in[26]};
  const float* lb[3]   = {(const float*)d_in[19], (const float*)d_in[23], (const float*)d_in[27]};
  const float* gg[3]   = {(const float*)d_in[20], (const float*)d_in[24], (const float*)d_in[28]};
  const float* bb[3]   = {(const float*)d_in[21], (const float*)d_in[25], (const float*)d_in[29]};

  // ---- workspace carve-up ----
  char* wsb = (char*)d_ws;
  size_t off = 0;
  auto alloc = [&](size_t bytes) -> void* {
    void* p = wsb + off;
    off += (bytes + 255) & ~(size_t)255;
    return p;
  };
  float*    bufA   = (float*)alloc((size_t)NNODES * HID * 4);   // hg
  float*    bufB   = (float*)alloc((size_t)NNODES * HID * 4);   // bn output
  float*    bufC   = (float*)alloc((size_t)NNODES * HID * 4);   // lin + agg / next h
  float*    colsum   = (float*)alloc(HID * 4);
  float*    colsumsq = (float*)alloc(HID * 4);
  float*    scaleB   = (float*)alloc(HID * 4);
  float*    shiftB   = (float*)alloc(HID * 4);
  float*    s_src  = (float*)alloc((size_t)NNODES * 4 * 4);
  float*    s_dst  = (float*)alloc((size_t)NNODES * 4 * 4);
  unsigned* m_enc  = (unsigned*)alloc((size_t)NNODES * 4 * 4);
  float*    denom  = (float*)alloc((size_t)NNODES * 4 * 4);
  float*    proj   = (float*)alloc(10 * 4 * 4);
  float*    logits = (float*)alloc((size_t)NEDGES * 4 * 4);
  float*    wbuf   = (float*)alloc((size_t)NEDGES * 4 * 4);

  struct Cfg { int fin, heads, O; };
  const Cfg cfg[3] = {{IN_F, 4, 256}, {HID, 4, 256}, {HID, 1, 2}};

  const float* h_in = x;
  for (int L = 0; L < 3; ++L) {
    const int fin = cfg[L].fin, heads = cfg[L].heads, O = cfg[L].O;
    const int HO = heads * O;
    float* lin_out = (L == 2) ? (float*)d_out : bufC;

    // ---- BatchNorm (batch stats) ----
    fill_f32_k<<<(fin + 255) / 256, 256, 0, stream>>>(colsum, 0.f, (size_t)fin);
    fill_f32_k<<<(fin + 255) / 256, 256, 0, stream>>>(colsumsq, 0.f, (size_t)fin);
    bn_reduce_k<<<dim3((fin + 255) / 256, (NNODES + 127) / 128), 256, 0, stream>>>(
        h_in, colsum, colsumsq, NNODES, fin);
    bn_scale_k<<<(fin + 255) / 256, 256, 0, stream>>>(
        colsum, colsumsq, gg[L], bb[L], scaleB, shiftB, fin, 1.0f / NNODES);
    size_t bn_tot = (size_t)NNODES * fin;
    bn_apply_k<<<(unsigned)((bn_tot + 255) / 256), 256, 0, stream>>>(
        h_in, scaleB, shiftB, bufB, bn_tot, fin);

    // ---- two GEMMs from BN output: hg = bn@Wg, lin = bn@lw + lb + cb ----
    if (HO >= 128) {
      dim3 grid(HO / 128, NNODES / 16);
      gemm_wmma_k<<<grid, 256, 0, stream>>>(bufB, Wg[L], nullptr, nullptr, bufA, NNODES, fin, HO);
      gemm_wmma_k<<<grid, 256, 0, stream>>>(bufB, lw[L], lb[L], cb[L], lin_out, NNODES, fin, HO);
    } else {
      gemv2_k<<<(NNODES + 7) / 8, 256, 0, stream>>>(bufB, Wg[L], nullptr, nullptr, bufA, NNODES, fin);
      gemv2_k<<<(NNODES + 7) / 8, 256, 0, stream>>>(bufB, lw[L], lb[L], cb[L], lin_out, NNODES, fin);
    }

    // ---- attention ----
    gat_scores_k<<<(NNODES * heads + 7) / 8, 256, 0, stream>>>(bufA, attn[L], s_src, s_dst,
                                                               NNODES, heads, O);
    gat_proj_k<<<(10 * heads + 7) / 8, 256, 0, stream>>>(We[L], attn[L], proj, heads, O);
    fill_u32_k<<<(NNODES * heads + 255) / 256, 256, 0, stream>>>(m_enc, ENC_NEG_INF,
                                                                 (size_t)NNODES * heads);
    fill_f32_k<<<(NNODES * heads + 255) / 256, 256, 0, stream>>>(denom, 0.f,
                                                                 (size_t)NNODES * heads);
    gat_logits_k<<<(NEDGES * heads + 255) / 256, 256, 0, stream>>>(
        ei, ea, s_src, s_dst, proj, at[L], logits, m_enc, NEDGES, heads);
    gat_softmax_k<<<(NEDGES * heads + 255) / 256, 256, 0, stream>>>(
        ei, logits, m_enc, wbuf, denom, NEDGES, heads);

    // ---- weighted message scatter into lin_out (holds lin + lb + cb) ----
    size_t mtot = (size_t)NEDGES * HO;
    gat_message_k<<<(unsigned)((mtot + 255) / 256), 256, 0, stream>>>(
        ei, ea, We[L], bufA, wbuf, denom, lin_out, NEDGES, heads, O);

    // ---- ReLU ----
    size_t rtot = (size_t)NNODES * HO;
    relu_k<<<(unsigned)((rtot + 255) / 256), 256, 0, stream>>>(lin_out, rtot);

    h_in = lin_out;
  }
}